// GCN_bnif_32238024523886
// MI455X (gfx1250) — compile-verified
//
#include <hip/hip_runtime.h>
#include <hip/hip_bf16.h>
#include <math.h>

typedef __attribute__((ext_vector_type(2))) float v2f;
typedef __attribute__((ext_vector_type(8))) float v8f;

#define EPS_BN 1e-5f

// ---------------- degree / norm ----------------
__global__ void k_init_deg(float* deg, int n) {
    int i = blockIdx.x * blockDim.x + threadIdx.x;
    if (i < n) deg[i] = 1.0f;  // self-loop contributes 1
}

__global__ void k_count_deg(float* deg, const int* __restrict__ dst, int e) {
    int i = blockIdx.x * blockDim.x + threadIdx.x;
    if (i < e) atomicAdd(&deg[dst[i]], 1.0f);
}

__global__ void k_rsqrt_inplace(float* d, int n) {
    int i = blockIdx.x * blockDim.x + threadIdx.x;
    if (i < n) {
        float v = d[i];
        d[i] = (v > 0.0f) ? rsqrtf(v) : 0.0f;
    }
}

// ---------------- WMMA GEMM + fused self-loop/bias epilogue ----------------
// H[M x NC] = A[M x 128] * B[128 x NC];  AGG = bias + dinv^2 * H
// B staged in LDS, pair-interleaved so each lane's (k,k+1) B pair is one ds_load_b64.
template <int NT>
__global__ void k_gemm_wmma(const float* __restrict__ A, const float* __restrict__ B,
                            float* __restrict__ H, float* __restrict__ AGG,
                            const float* __restrict__ dinv, const float* __restrict__ bias,
                            int M) {
    constexpr int NC = NT * 16;
    constexpr int LOG_NC = (NT == 8) ? 7 : 6;
    __shared__ float ldsB[128 * NC];  // 64KB (NT=8) / 32KB (NT=4); WGP has 320KB

    // cooperative stage of B: ldsB[(k>>1)*(2*NC) + 2*n + (k&1)] = B[k*NC + n]
    for (int idx = threadIdx.x; idx < 128 * NC; idx += blockDim.x) {
        int k = idx >> LOG_NC;
        int n = idx & (NC - 1);
        ldsB[((k >> 1) << (LOG_NC + 1)) + 2 * n + (k & 1)] = B[idx];
    }
    __syncthreads();

    int lane = threadIdx.x & 31;
    int wave = threadIdx.x >> 5;
    int m0 = (blockIdx.x * (blockDim.x >> 5) + wave) * 16;
    if (m0 >= M) return;  // wave-uniform: EXEC stays all-ones for WMMA

    int half = lane >> 4;   // 0: lanes 0-15, 1: lanes 16-31
    int mr = lane & 15;

    int arow = m0 + mr;
    if (arow >= M) arow = M - 1;  // clamp (garbage rows never stored)
    const float* Arow = A + (size_t)arow * 128;

    v8f acc[NT];
#pragma unroll
    for (int t = 0; t < NT; ++t) acc[t] = (v8f){0.f, 0.f, 0.f, 0.f, 0.f, 0.f, 0.f, 0.f};

    for (int k0 = 0; k0 < 128; k0 += 4) {
        // A 16x4 fp32 layout: lane half selects K pair {0,1} / {2,3}
        v2f a = *(const v2f*)(Arow + k0 + 2 * half);
        int p = (k0 >> 1) + half;  // B K-pair index
        const float* bbase = &ldsB[(p << (LOG_NC + 1)) + 2 * mr];
#pragma unroll
        for (int t = 0; t < NT; ++t) {
            v2f b = *(const v2f*)(bbase + t * 32);  // ds_load_b64, const offset per tile
            acc[t] = __builtin_amdgcn_wmma_f32_16x16x4_f32(
                false, a, false, b, (short)0, acc[t], false, false);
        }
    }

    // epilogue: H = acc; AGG = bias + dinv^2 * acc
    // C 16x16 fp32 layout: VGPR r -> row m0 + r + 8*half, col = tile*16 + mr
    float di2[8];
#pragma unroll
    for (int r = 0; r < 8; ++r) {
        int row = m0 + r + 8 * half;
        float di = (row < M) ? dinv[row] : 0.0f;
        di2[r] = di * di;
    }
#pragma unroll
    for (int t = 0; t < NT; ++t) {
        float bc = bias[t * 16 + mr];
#pragma unroll
        for (int r = 0; r < 8; ++r) {
            int row = m0 + r + 8 * half;
            if (row < M) {
                float v = acc[t][r];
                size_t off = (size_t)row * NC + t * 16 + mr;
                H[off] = v;
                AGG[off] = bc + di2[r] * v;
            }
        }
    }
}

// ---------------- edge scatter: wave32 per edge, contiguous chunk per lane ----------------
template <int C>
__global__ void k_scatter(const float* __restrict__ h, const int* __restrict__ src,
                          const int* __restrict__ dst, const float* __restrict__ dinv,
                          float* __restrict__ agg, int e) {
    constexpr int CHUNK = C / 32;
    int warp = (blockIdx.x * blockDim.x + threadIdx.x) >> 5;
    int lane = threadIdx.x & 31;
    if (warp >= e) return;
    int s = src[warp];
    int d = dst[warp];
    float nrm = dinv[s] * dinv[d];
    const float* hs = h + (size_t)s * C + lane * CHUNK;
    float* ad = agg + (size_t)d * C + lane * CHUNK;
    float v[CHUNK];
#pragma unroll
    for (int j = 0; j < CHUNK; ++j) v[j] = hs[j];  // merges to b128/b64
#pragma unroll
    for (int j = 0; j < CHUNK; ++j) atomicAdd(&ad[j], nrm * v[j]);
}

// ---------------- BN (eval) + ReLU, out of place (agg -> act) ----------------
__global__ void k_bn_relu(const float* __restrict__ x, float* __restrict__ y,
                          const float* __restrict__ gamma, const float* __restrict__ beta,
                          const float* __restrict__ rm, const float* __restrict__ rv,
                          int total, int cmask) {
    int i = blockIdx.x * blockDim.x + threadIdx.x;
    if (i < total) {
        int c = i & cmask;
        float v = (x[i] - rm[c]) * rsqrtf(rv[c] + EPS_BN) * gamma[c] + beta[c];
        y[i] = fmaxf(v, 0.0f);
    }
}

// ---------------- log_softmax over 64 channels: wave32 per node ----------------
__global__ void k_log_softmax64(const float* __restrict__ x, float* __restrict__ out, int n) {
    int warp = (blockIdx.x * blockDim.x + threadIdx.x) >> 5;
    int lane = threadIdx.x & 31;
    if (warp >= n) return;
    const float* xp = x + (size_t)warp * 64;
    float v0 = xp[lane];
    float v1 = xp[lane + 32];
    float m = fmaxf(v0, v1);
#pragma unroll
    for (int off = 16; off >= 1; off >>= 1) m = fmaxf(m, __shfl_xor(m, off, 32));
    float s = __expf(v0 - m) + __expf(v1 - m);
#pragma unroll
    for (int off = 16; off >= 1; off >>= 1) s += __shfl_xor(s, off, 32);
    float lse = m + __logf(s);
    float* op = out + (size_t)warp * 64;
    op[lane] = v0 - lse;
    op[lane + 32] = v1 - lse;
}

extern "C" void kernel_launch(void* const* d_in, const int* in_sizes, int n_in,
                              void* d_out, int out_size, void* d_ws, size_t ws_size,
                              hipStream_t stream) {
    const float* x      = (const float*)d_in[0];
    const int*   ei     = (const int*)d_in[1];
    const float* W1     = (const float*)d_in[2];
    const float* b1     = (const float*)d_in[3];
    const float* gamma1 = (const float*)d_in[4];
    const float* beta1  = (const float*)d_in[5];
    const float* rm1    = (const float*)d_in[6];
    const float* rv1    = (const float*)d_in[7];
    const float* W2     = (const float*)d_in[8];
    const float* b2     = (const float*)d_in[9];
    const float* gamma2 = (const float*)d_in[10];
    const float* beta2  = (const float*)d_in[11];
    const float* rm2    = (const float*)d_in[12];
    const float* rv2    = (const float*)d_in[13];
    const float* W3     = (const float*)d_in[14];
    const float* b3     = (const float*)d_in[15];
    float* out = (float*)d_out;

    const int Nn = in_sizes[0] / 128;  // 100000
    const int E  = in_sizes[1] / 2;    // 1600000
    const int* src = ei;
    const int* dst = ei + E;

    // workspace: dinv | bufH | bufA | bufX  (ping-pong so fused epilogue never
    // overwrites a buffer other blocks are still reading)
    float* dinv = (float*)d_ws;                 // Nn
    float* bufH = dinv + Nn;                    // Nn*128 (GEMM output, gathered by edges)
    float* bufA = bufH + (size_t)Nn * 128;      // Nn*128 (scatter accumulator)
    float* bufX = bufA + (size_t)Nn * 128;      // Nn*128 (activation / GEMM input)

    const int TB = 256;
    const int nodeBlocks = (Nn + TB - 1) / TB;
    const int edgeBlocks = (E + TB - 1) / TB;
    const int ewBlocks   = (E + (TB / 32) - 1) / (TB / 32);     // wave-per-edge
    const int nwBlocks   = (Nn + (TB / 32) - 1) / (TB / 32);    // wave-per-node
    const int el128      = Nn * 128;
    const int b128       = (el128 + TB - 1) / TB;
    const int rowBlocks  = (Nn + 15) / 16;
    const int gemmBlocks = (rowBlocks + (TB / 32) - 1) / (TB / 32);

    // ---- symmetric GCN normalization: dinv = rsqrt(1 + in-degree) ----
    k_init_deg<<<nodeBlocks, TB, 0, stream>>>(dinv, Nn);
    k_count_deg<<<edgeBlocks, TB, 0, stream>>>(dinv, dst, E);
    k_rsqrt_inplace<<<nodeBlocks, TB, 0, stream>>>(dinv, Nn);

    // ---- layer 1: conv(x, W1) -> BN -> ReLU ----
    k_gemm_wmma<8><<<gemmBlocks, TB, 0, stream>>>(x, W1, bufH, bufA, dinv, b1, Nn);
    k_scatter<128><<<ewBlocks, TB, 0, stream>>>(bufH, src, dst, dinv, bufA, E);
    k_bn_relu<<<b128, TB, 0, stream>>>(bufA, bufX, gamma1, beta1, rm1, rv1, el128, 127);

    // ---- layer 2 ----
    k_gemm_wmma<8><<<gemmBlocks, TB, 0, stream>>>(bufX, W2, bufH, bufA, dinv, b2, Nn);
    k_scatter<128><<<ewBlocks, TB, 0, stream>>>(bufH, src, dst, dinv, bufA, E);
    k_bn_relu<<<b128, TB, 0, stream>>>(bufA, bufX, gamma2, beta2, rm2, rv2, el128, 127);

    // ---- output layer + log_softmax ----
    k_gemm_wmma<4><<<gemmBlocks, TB, 0, stream>>>(bufX, W3, bufH, bufA, dinv, b3, Nn);
    k_scatter<64><<<ewBlocks, TB, 0, stream>>>(bufH, src, dst, dinv, bufA, E);
    k_log_softmax64<<<nwBlocks, TB, 0, stream>>>(bufA, out, Nn);
}